// MultipleTopoTreeLoss_28484223107629
// MI455X (gfx1250) — compile-verified
//
#include <hip/hip_runtime.h>
#include <math.h>

// ---- CDNA5 types ----
typedef float v2f __attribute__((ext_vector_type(2)));
typedef float v8f __attribute__((ext_vector_type(8)));

#define N_PTS   1024
#define THREADS 256
#define PPT     4          // points owned per thread (256*4 = 1024)
#define NPROB   32         // 16 samples x 2 chunks
#define EPS_F   1e-8f

__device__ __forceinline__ unsigned long long minu64(unsigned long long a,
                                                     unsigned long long b) {
  return a < b ? a : b;
}

// 64-bit xor-shuffle built from two 32-bit lane shuffles (wave32)
__device__ __forceinline__ unsigned long long shflxor64(unsigned long long v, int m) {
  unsigned lo = (unsigned)v;
  unsigned hi = (unsigned)(v >> 32);
  lo = __shfl_xor(lo, m, 32);
  hi = __shfl_xor(hi, m, 32);
  return ((unsigned long long)hi << 32) | lo;
}

__global__ __launch_bounds__(THREADS)
void topo_mst_kernel(const float* __restrict__ r, float* __restrict__ partial) {
  __shared__ float xs[N_PTS];
  __shared__ float ys[N_PTS];
  __shared__ float zs[N_PTS];
  __shared__ unsigned long long redK[8];
  __shared__ float stat_mu[3];
  __shared__ float stat_inv[3];

  const int tid   = threadIdx.x;
  const int lane  = tid & 31;
  const int wave  = tid >> 5;
  const int b     = blockIdx.x;   // 0..31
  const int samp  = b >> 1;       // 0..15
  const int chunk = b & 1;        // 0..1

  const float* src = r + (size_t)((samp * (2 * N_PTS) + chunk * N_PTS) * 3);

  // ---- stage the 1024x3 chunk into LDS (SoA layout), one HBM touch ----
  __builtin_prefetch(src + tid, 0, 1);                 // global_prefetch_b8
  for (int t = tid; t < N_PTS * 3; t += THREADS) {
    int p  = t / 3;
    int ch = t - 3 * p;
    float v = src[t];
    if (ch == 0) xs[p] = v; else if (ch == 1) ys[p] = v; else zs[p] = v;
  }
  __syncthreads();

  // ---- per-channel moments (sum, sum of squares) on the matrix pipe ----
  // A tile t (16x4 f32) = 64 consecutive channel values. ISA layout:
  //   VGPR0: lanes0-15 K=0 (=val[4m+0]), lanes16-31 K=2 (=val[4m+2])
  //   VGPR1: lanes0-15 K=1,              lanes16-31 K=3
  // For B = A^T (B[k][n]=val[4n+k]) the register image is identical to A,
  // so csq = sum_t A_t * A_t^T gives trace = sum(x^2); csum = sum_t A_t * ones
  // gives row sums (column-replicated) = sum(x) pieces.
  if (wave < 3) {
    const float* v = (wave == 0) ? xs : (wave == 1) ? ys : zs;
    v8f csum = {};
    v8f csq  = {};
    v2f ones; ones.x = 1.0f; ones.y = 1.0f;
    const int row  = lane & 15;
    const int koff = (lane < 16) ? 0 : 2;
    for (int t = 0; t < 16; ++t) {
      const int base = t * 64 + row * 4 + koff;
      v2f a; a.x = v[base]; a.y = v[base + 1];
      csum = __builtin_amdgcn_wmma_f32_16x16x4_f32(
          false, a, false, ones, (short)0, csum, false, false);
      csq  = __builtin_amdgcn_wmma_f32_16x16x4_f32(
          false, a, false, a,    (short)0, csq,  false, false);
    }
    float S = 0.0f, Q = 0.0f;
#pragma unroll
    for (int vg = 0; vg < 8; ++vg) {
      // C/D layout: VGPR vg holds M=vg (lanes0-15, N=lane) and M=vg+8
      // (lanes16-31, N=lane-16).
      S += __shfl(csum[vg], 0, 32)  + __shfl(csum[vg], 16, 32);      // N=0 col
      Q += __shfl(csq[vg],  vg, 32) + __shfl(csq[vg], 24 + vg, 32);  // diagonal
    }
    if (lane == 0) {
      const float mu  = S * (1.0f / N_PTS);
      const float var = fmaxf(Q * (1.0f / N_PTS) - mu * mu, 0.0f);
      stat_mu[wave]  = mu;
      stat_inv[wave] = 1.0f / (sqrtf(var) + EPS_F);
    }
  }
  __syncthreads();

  // ---- normalize in place ----
  {
    const float mx = stat_mu[0], ix = stat_inv[0];
    const float my = stat_mu[1], iy = stat_inv[1];
    const float mz = stat_mu[2], iz = stat_inv[2];
    for (int p = tid; p < N_PTS; p += THREADS) {
      xs[p] = (xs[p] - mx) * ix;
      ys[p] = (ys[p] - my) * iy;
      zs[p] = (zs[p] - mz) * iz;
    }
  }
  __syncthreads();

  // ---- Prim's MST: register-resident owned points + mind, LDS argmin ----
  const float INF = __int_as_float(0x7f800000);
  float px[PPT], py[PPT], pz[PPT], mind[PPT];
  bool  vis[PPT];
  const float p0x = xs[0], p0y = ys[0], p0z = zs[0];
#pragma unroll
  for (int k = 0; k < PPT; ++k) {
    const int i = tid + k * THREADS;
    px[k] = xs[i]; py[k] = ys[i]; pz[k] = zs[i];
    const float dx = px[k] - p0x, dy = py[k] - p0y, dz = pz[k] - p0z;
    const float d  = dx * dx + dy * dy + dz * dz;
    vis[k]  = (i == 0);
    mind[k] = vis[k] ? INF : d;
  }

  float total = 0.0f;
  for (int step = 0; step < N_PTS - 1; ++step) {
    // local candidate: pack (dist_bits << 32) | index; non-negative floats
    // order identically to their bit patterns, low index wins ties (matches
    // jnp.argmin first-min semantics).
    unsigned long long key = ~0ull;
#pragma unroll
    for (int k = 0; k < PPT; ++k) {
      const unsigned long long kk =
          ((unsigned long long)__float_as_uint(mind[k]) << 32) |
          (unsigned)(tid + k * THREADS);
      key = minu64(key, kk);
    }
    // wave32 reduction
#pragma unroll
    for (int off = 16; off > 0; off >>= 1)
      key = minu64(key, shflxor64(key, off));
    if (lane == 0) redK[wave] = key;
    __syncthreads();
    unsigned long long best = redK[0];
#pragma unroll
    for (int w = 1; w < 8; ++w) best = minu64(best, redK[w]);

    const int   j    = (int)(best & 0xffffffffu);
    total += __uint_as_float((unsigned)(best >> 32));

    const float pjx = xs[j], pjy = ys[j], pjz = zs[j];
#pragma unroll
    for (int k = 0; k < PPT; ++k) {
      const int i = tid + k * THREADS;
      if (i == j) {
        vis[k] = true; mind[k] = INF;
      } else if (!vis[k]) {
        const float dx = px[k] - pjx, dy = py[k] - pjy, dz = pz[k] - pjz;
        const float d  = dx * dx + dy * dy + dz * dz;
        mind[k] = fminf(mind[k], d);
      }
    }
    __syncthreads();   // protect redK reuse next iteration
  }

  if (tid == 0) partial[b] = total;
}

// Deterministic fixed-order final reduction: mean over chunks of mean over
// batch with equal sizes == sum(all 32) / 32.
__global__ void topo_finalize_kernel(const float* __restrict__ partial,
                                     float* __restrict__ out) {
  if (blockIdx.x == 0 && threadIdx.x == 0) {
    float s = 0.0f;
    for (int i = 0; i < NPROB; ++i) s += partial[i];
    out[0] = s * (1.0f / (float)NPROB);
  }
}

extern "C" void kernel_launch(void* const* d_in, const int* in_sizes, int n_in,
                              void* d_out, int out_size, void* d_ws, size_t ws_size,
                              hipStream_t stream) {
  (void)in_sizes; (void)n_in; (void)out_size; (void)ws_size;
  const float* r       = (const float*)d_in[0];   // [16, 2048, 3] f32
  float*       partial = (float*)d_ws;            // 32 floats of scratch
  float*       out     = (float*)d_out;           // scalar f32

  topo_mst_kernel<<<dim3(NPROB), dim3(THREADS), 0, stream>>>(r, partial);
  topo_finalize_kernel<<<dim3(1), dim3(32), 0, stream>>>(partial, out);
}